// HungarianMatcherDynamicK_49220325212496
// MI455X (gfx1250) — compile-verified
//
#include <hip/hip_runtime.h>
#include <cstdint>

// Problem constants (from reference setup_inputs / module constants)
constexpr int B    = 8;
constexpr int NP   = 5000;   // predictions per batch
constexpr int NG   = 300;    // ground-truth boxes per batch
constexpr int K    = 5;      // TOPK
constexpr int TM   = 4;      // gt boxes per block
constexpr int TPB  = 256;    // threads per block (8 wave32)
constexpr int NWAVE = TPB / 32;
constexpr int NTILE = (NP + TPB - 1) / TPB;  // 20 pred tiles

// ---------------------------------------------------------------------------
// CDNA5 async global->LDS copy (ASYNCcnt-tracked DMA, no VGPR staging).
// LDS destination = byte offset in VGPR lds_off; global addr = SGPR64 base + VGPR voff.
// ---------------------------------------------------------------------------
__device__ __forceinline__ void async_load_b128(unsigned lds_off, unsigned voff,
                                                unsigned long long base) {
  asm volatile("global_load_async_to_lds_b128 %0, %1, %2"
               :: "v"(lds_off), "v"(voff), "s"(base)
               : "memory");
}

__device__ __forceinline__ void wait_asynccnt_le1() {
  asm volatile("s_wait_asynccnt 1" ::: "memory");
}
__device__ __forceinline__ void wait_asynccnt_0() {
  asm volatile("s_wait_asynccnt 0" ::: "memory");
}

// Flat pointer to __shared__ carries the LDS byte offset in its low 32 bits
// (aperture lives in addr[63:32], ISA 10.2).
__device__ __forceinline__ unsigned lds_addr_of(const void* p) {
  return (unsigned)(uintptr_t)p;
}

__device__ __forceinline__ unsigned long long u64min(unsigned long long a,
                                                     unsigned long long b) {
  return (b < a) ? b : a;
}

__global__ __launch_bounds__(TPB)
void matcher_topk_kernel(const float* __restrict__ pred_box,
                         const float* __restrict__ gt_box,
                         int* __restrict__ out) {
  __shared__ float4              tile[2][TPB];          // 8 KB double buffer
  __shared__ unsigned long long  keybuf[TPB * K];       // 10 KB candidates
  __shared__ unsigned long long  part[NWAVE];           // 64 B wave partials

  const int tid  = threadIdx.x;
  const int lane = tid & 31;
  const int wid  = tid >> 5;
  const int b    = blockIdx.y;
  const int g0   = blockIdx.x * TM;

  // ---- GT boxes for this block: uniform -> one s_load_b512 ----
  float gx0[TM], gy0[TM], gx1[TM], gy1[TM], garea[TM];
  const float* gp = gt_box + ((size_t)b * NG + g0) * 4;
#pragma unroll
  for (int i = 0; i < TM; ++i) {
    gx0[i] = gp[4 * i + 0];
    gy0[i] = gp[4 * i + 1];
    gx1[i] = gp[4 * i + 2];
    gy1[i] = gp[4 * i + 3];
    garea[i] = (gx1[i] - gx0[i]) * (gy1[i] - gy0[i]);
  }

  // ---- per-thread top-5 lists, packed key = (cost_bits<<32) | pred_idx ----
  // All indices below are compile-time constants -> stays in VGPRs.
  unsigned long long lst[TM][K];
#pragma unroll
  for (int i = 0; i < TM; ++i)
#pragma unroll
    for (int k = 0; k < K; ++k) lst[i][k] = ~0ull;

  const unsigned long long base =
      (unsigned long long)(uintptr_t)(pred_box + (size_t)b * NP * 4);
  const unsigned lds0 = lds_addr_of(&tile[0][tid]);
  const unsigned lds1 = lds_addr_of(&tile[1][tid]);

  // ---- async pipeline prologue: tile 0 ----
  {
    int j0 = tid < NP ? tid : NP - 1;
    async_load_b128(lds0, (unsigned)(j0 * 16), base);
  }

#pragma unroll 1
  for (int t = 0; t < NTILE; ++t) {
    if (t + 1 < NTILE) {
      int jn = (t + 1) * TPB + tid;
      if (jn >= NP) jn = NP - 1;                 // clamp OOB lanes
      async_load_b128((t & 1) ? lds0 : lds1, (unsigned)(jn * 16), base);
      wait_asynccnt_le1();                        // oldest (tile t) done
    } else {
      wait_asynccnt_0();
    }

    const float4 p = tile[t & 1][tid];            // ds_load_b128, lane-private
    const int j = t * TPB + tid;

    const float px0 = p.x, py0 = p.y, px1 = p.z, py1 = p.w;
    const float area_p = (px1 - px0) * (py1 - py0);

#pragma unroll
    for (int i = 0; i < TM; ++i) {
      float l1 = fabsf(px0 - gx0[i]) + fabsf(py0 - gy0[i]) +
                 fabsf(px1 - gx1[i]) + fabsf(py1 - gy1[i]);
      // intersection
      float ix0 = fmaxf(px0, gx0[i]), iy0 = fmaxf(py0, gy0[i]);
      float ix1 = fminf(px1, gx1[i]), iy1 = fminf(py1, gy1[i]);
      float iw = fmaxf(ix1 - ix0, 0.0f), ih = fmaxf(iy1 - iy0, 0.0f);
      float inter = iw * ih;
      float uni = area_p + garea[i] - inter;
      // enclosing box
      float ex0 = fminf(px0, gx0[i]), ey0 = fminf(py0, gy0[i]);
      float ex1 = fmaxf(px1, gx1[i]), ey1 = fmaxf(py1, gy1[i]);
      float ae = (ex1 - ex0) * (ey1 - ey0);
      // Ranking cost: 1-ulp v_rcp_f32 instead of IEEE divide (saves ~20
      // VALU ops/pair of div_scale/div_fmas/div_fixup expansion).
      float giou = inter * __builtin_amdgcn_rcpf(uni) -
                   (ae - uni) * __builtin_amdgcn_rcpf(ae);
      float cost = l1 + 1.0f - giou;              // >= 0 -> bits are monotonic

      unsigned long long key =
          ((unsigned long long)__float_as_uint(cost) << 32) | (unsigned)j;
      if (j < NP && key < lst[i][K - 1]) {
        lst[i][K - 1] = key;
#pragma unroll
        for (int q = K - 1; q > 0; --q) {
          unsigned long long a = lst[i][q - 1], c = lst[i][q];
          lst[i][q - 1] = (c < a) ? c : a;
          lst[i][q]     = (c < a) ? a : c;
        }
      }
    }
  }

  __syncthreads();

  // ---- block-wide top-5 per GT (2 barriers per extraction pass) ----
#pragma unroll
  for (int i = 0; i < TM; ++i) {
#pragma unroll
    for (int k = 0; k < K; ++k) keybuf[tid * K + k] = lst[i][k];
    __syncthreads();

#pragma unroll 1
    for (int pass = 0; pass < K; ++pass) {
      // strided min over this thread's stripe of the 1280 candidates
      unsigned long long m = keybuf[tid];
#pragma unroll
      for (int e = 1; e < K; ++e) m = u64min(m, keybuf[e * TPB + tid]);

      // wave32 min via lane shuffles (ds_permute path, no barriers)
#pragma unroll
      for (int off = 16; off > 0; off >>= 1)
        m = u64min(m, __shfl_xor(m, off, 32));
      if (lane == 0) part[wid] = m;
      __syncthreads();

      // every thread redundantly reduces the 8 wave partials -> no broadcast
      unsigned long long best = part[0];
#pragma unroll
      for (int w = 1; w < NWAVE; ++w) best = u64min(best, part[w]);

      if (tid == 0) {
        out[((size_t)b * NG + (g0 + i)) * K + pass] =
            (int)(unsigned)(best & 0xffffffffu);
      }
      // pred indices are globally unique across candidates -> blank by key
#pragma unroll
      for (int k = 0; k < K; ++k) {
        if (keybuf[tid * K + k] == best) keybuf[tid * K + k] = ~0ull;
      }
      __syncthreads();   // also protects part[] reuse next pass
    }
  }

  // ---- matched_gt (second output, concatenated after matched_pred) ----
  if (tid < TM * K) {
    const int i = tid / K;
    const int k = tid % K;
    out[(size_t)B * NG * K + ((size_t)b * NG + (g0 + i)) * K + k] = g0 + i;
  }
}

extern "C" void kernel_launch(void* const* d_in, const int* in_sizes, int n_in,
                              void* d_out, int out_size, void* d_ws, size_t ws_size,
                              hipStream_t stream) {
  // setup_inputs order: pred_box, pred_obj, gt_box, gt_obj
  const float* pred_box = (const float*)d_in[0];
  const float* gt_box   = (const float*)d_in[2];
  int* out = (int*)d_out;  // int32 outputs: matched_pred | matched_gt

  dim3 grid(NG / TM, B);   // (75, 8) blocks
  matcher_topk_kernel<<<grid, TPB, 0, stream>>>(pred_box, gt_box, out);
}